// CMECell_1211180777609
// MI455X (gfx1250) — compile-verified
//
#include <hip/hip_runtime.h>
#include <cstdint>

#define TSTEPS 4096
#define DCH 64
#define NT 50
#define NK 10
#define NCHAIN (DCH * NT) /* 3200 */
#define CH_MAX 2048

#if defined(__gfx1250__) && __has_builtin(__builtin_amdgcn_tensor_load_to_lds) && \
    __has_builtin(__builtin_amdgcn_s_wait_tensorcnt)
#define USE_TDM 1
#else
#define USE_TDM 0
#endif

typedef unsigned int v4u __attribute__((ext_vector_type(4)));
typedef int v8i __attribute__((ext_vector_type(8)));
typedef int v4i __attribute__((ext_vector_type(4)));

#if USE_TDM
// DMA one strided column (stride = DCH floats) of `rows` fp32 elements into LDS.
// D# group0: count=1, lds_addr, 57-bit global_addr, type=2 ("image").
// D# group1: wg_mask=0, data_size=2 (4B), tensor = DCH x TSTEPS,
//            tile = 1 x rows, dim0_stride = DCH, dim1_stride = DCH.
__device__ __forceinline__ void tdm_load_column(const float* gsrc, float* ldsDst, int rows) {
  unsigned lds = (unsigned)(unsigned long long)(uintptr_t)ldsDst; // low 32 bits = LDS offset
  unsigned long long ga = (unsigned long long)(uintptr_t)gsrc;
  v4u g0 = {1u, lds, (unsigned)(ga & 0xFFFFFFFFull),
            (unsigned)(((ga >> 32) & 0x1FFFFFFull) | 0x80000000u)};
  v8i g1;
  g1[0] = 0x00020000;                                             // data_size=4B
  g1[1] = (int)((DCH & 0xFFFF) << 16);                            // tensor_dim0 lo16 @bit48
  g1[2] = (int)(((DCH >> 16) & 0xFFFF) | ((TSTEPS & 0xFFFF) << 16)); // dim0 hi | dim1 lo
  g1[3] = (int)(((TSTEPS >> 16) & 0xFFFF) | (1 << 16));           // dim1 hi | tile_dim0=1
  g1[4] = (int)(rows & 0xFFFF);                                   // tile_dim1 | tile_dim2=0
  g1[5] = DCH;                                                    // tensor_dim0_stride lo32
  g1[6] = (int)((DCH & 0xFFFF) << 16);                            // stride0 hi=0 | stride1 lo16
  g1[7] = 0;                                                      // stride1 bits 47:16
  v4i g2 = {0, 0, 0, 0};
  v4i g3 = {0, 0, 0, 0};
  v8i g4 = {0, 0, 0, 0, 0, 0, 0, 0}; // extra group on 6-arg toolchain; zero-filled
  __builtin_amdgcn_tensor_load_to_lds(g0, g1, g2, g3, g4, 0);
}
#endif

// Stage f[t0..t0+rows) and alpha[t0..t0+rows) for a fixed channel d into LDS.
__device__ __forceinline__ void stage_tile(const float* __restrict__ f,
                                           const float* __restrict__ alpha,
                                           float* sf, float* sal, size_t elemOff, int rows) {
#if USE_TDM
  int wid = (int)(threadIdx.x >> 5); // uniform per wave32
  if (wid == 0)
    tdm_load_column(f + elemOff, sf, rows);
  else if (wid == 1)
    tdm_load_column(alpha + elemOff, sal, rows);
  __builtin_amdgcn_s_wait_tensorcnt(0);
  __syncthreads();
#else
  for (int i = (int)threadIdx.x; i < rows; i += (int)blockDim.x) {
    sf[i] = f[elemOff + (size_t)i * DCH];
    sal[i] = alpha[elemOff + (size_t)i * DCH];
  }
  __syncthreads();
#endif
}

// Per-step coefficients exploiting s[tau,k] = (mu1/tau)*(1 + i*k*omega):
//   z_k = -(x + i*k*th);  hh_k = E * R^k (one exp + one sincos, rest complex muls)
//   ih_k = exprel(z_k) = (hh_k - 1)/z_k with |z|^2 < 1e-12 small-limit (matches reference).
__device__ __forceinline__ void step_coeffs(float x, float E, float th, float cs, float sn,
                                            float (&hr)[NK], float (&hi)[NK],
                                            float (&pr)[NK], float (&pi)[NK]) {
  float cr = E, ci = 0.0f;
  float x2 = x * x;
#pragma unroll
  for (int k = 0; k < NK; ++k) {
    hr[k] = cr;
    hi[k] = ci;
    float zi = -(float)k * th;
    float m2 = x2 + zi * zi;
    float nr = cr - 1.0f;
    float ni = ci;
    if (m2 < 1e-12f) {
      pr[k] = 1.0f - 0.5f * x;
      pi[k] = 0.5f * zi;
    } else {
      float rm = __builtin_amdgcn_rcpf(m2);
      pr[k] = (ni * zi - nr * x) * rm;
      pi[k] = (-ni * x - nr * zi) * rm;
    }
    float t1 = cr * cs + ci * sn; // cur *= (cs - i*sn)
    ci = ci * cs - cr * sn;
    cr = t1;
  }
}

// Pass 1: per (d, chunk) block, lane = tau. Compute chunk transfer (A = prod hh, B = zero-init run).
__global__ __launch_bounds__(64) void cme_pass1(const float* __restrict__ f,
                                                const float* __restrict__ alpha,
                                                const float2* __restrict__ s,
                                                float* __restrict__ ws1, int chunkLen) {
  __shared__ float sf[CH_MAX];
  __shared__ float sal[CH_MAX];
  int d = (int)blockIdx.x, c = (int)blockIdx.y;
  int tau = (int)threadIdx.x;
  int tt = tau < NT ? tau : NT - 1;
  float sr0 = s[tt * NK + 0].x;
  float si1 = s[tt * NK + 1].y;
  int t0 = c * chunkLen;
  stage_tile(f, alpha, sf, sal, (size_t)t0 * DCH + (size_t)d, chunkLen);

  float Ar[NK], Ai[NK], Br[NK], Bi[NK];
#pragma unroll
  for (int k = 0; k < NK; ++k) {
    Ar[k] = 1.0f; Ai[k] = 0.0f; Br[k] = 0.0f; Bi[k] = 0.0f;
  }
  for (int i = 0; i < chunkLen; ++i) {
    float ft = sf[i], at = sal[i];
    float x = at * sr0;
    float th = at * si1;
    float E = __expf(-x);
    float sn, cs;
    __sincosf(th, &sn, &cs);
    float hr[NK], hi[NK], pr[NK], pi[NK];
    step_coeffs(x, E, th, cs, sn, hr, hi, pr, pi);
#pragma unroll
    for (int k = 0; k < NK; ++k) {
      float br = Br[k] * hr[k] - Bi[k] * hi[k] + ft * pr[k];
      float bi = Br[k] * hi[k] + Bi[k] * hr[k] + ft * pi[k];
      Br[k] = br; Bi[k] = bi;
      float ar = Ar[k] * hr[k] - Ai[k] * hi[k];
      float ai = Ar[k] * hi[k] + Ai[k] * hr[k];
      Ar[k] = ar; Ai[k] = ai;
    }
  }
  if (tau < NT) {
    size_t chain = (size_t)d * NT + (size_t)tau;
    size_t base = ((size_t)c * NCHAIN + chain) * NK;
#pragma unroll
    for (int k = 0; k < NK; ++k) {
      ws1[(base + k) * 4 + 0] = Ar[k];
      ws1[(base + k) * 4 + 1] = Ai[k];
      ws1[(base + k) * 4 + 2] = Br[k];
      ws1[(base + k) * 4 + 3] = Bi[k];
    }
  }
}

// Pass 2: serial prefix over chunks per chain (3200 lanes). Stores F at each chunk start into ws2.
__global__ __launch_bounds__(64) void cme_pass2(const float2* __restrict__ F0,
                                                const float* __restrict__ ws1,
                                                float* __restrict__ ws2, int C) {
  int chain = (int)blockIdx.x * 64 + (int)threadIdx.x; // grid = 50 -> exactly 3200
  float Fr[NK], Fi[NK];
#pragma unroll
  for (int k = 0; k < NK; ++k) {
    float2 v = F0[(size_t)chain * NK + k];
    Fr[k] = v.x; Fi[k] = v.y;
  }
  for (int c = 0; c < C; ++c) {
    size_t base = ((size_t)c * NCHAIN + (size_t)chain) * NK;
#pragma unroll
    for (int k = 0; k < NK; ++k) {
      float ar = ws1[(base + k) * 4 + 0];
      float ai = ws1[(base + k) * 4 + 1];
      float br = ws1[(base + k) * 4 + 2];
      float bi = ws1[(base + k) * 4 + 3];
      ws2[(base + k) * 2 + 0] = Fr[k];
      ws2[(base + k) * 2 + 1] = Fi[k];
      float fr = Fr[k] * ar - Fi[k] * ai + br;
      float fi = Fr[k] * ai + Fi[k] * ar + bi;
      Fr[k] = fr; Fi[k] = fi;
    }
  }
}

// Pass 3: replay each chunk from its known initial state, emitting til outputs.
__global__ __launch_bounds__(64) void cme_pass3(const float* __restrict__ f,
                                                const float* __restrict__ alpha,
                                                const float2* __restrict__ s,
                                                const float2* __restrict__ eta,
                                                const float* __restrict__ tau_stars,
                                                const float* __restrict__ ws2,
                                                float* __restrict__ out, int chunkLen) {
  __shared__ float sf[CH_MAX];
  __shared__ float sal[CH_MAX];
  int d = (int)blockIdx.x, c = (int)blockIdx.y;
  int tau = (int)threadIdx.x;
  int tt = tau < NT ? tau : NT - 1;
  float sr0 = s[tt * NK + 0].x;
  float si1 = s[tt * NK + 1].y;
  float inv_tau = 1.0f / tau_stars[tt];
  float er[NK], ei[NK];
#pragma unroll
  for (int k = 0; k < NK; ++k) {
    float2 e = eta[k];
    er[k] = e.x; ei[k] = e.y;
  }
  int t0 = c * chunkLen;
  stage_tile(f, alpha, sf, sal, (size_t)t0 * DCH + (size_t)d, chunkLen);

  size_t chain = (size_t)d * NT + (size_t)tt;
  size_t base = ((size_t)c * NCHAIN + chain) * NK;
  float Fr[NK], Fi[NK];
#pragma unroll
  for (int k = 0; k < NK; ++k) {
    Fr[k] = ws2[(base + k) * 2 + 0];
    Fi[k] = ws2[(base + k) * 2 + 1];
  }
  for (int i = 0; i < chunkLen; ++i) {
    float ft = sf[i], at = sal[i];
    float x = at * sr0;
    float th = at * si1;
    float E = __expf(-x);
    float sn, cs;
    __sincosf(th, &sn, &cs);
    float hr[NK], hi[NK], pr[NK], pi[NK];
    step_coeffs(x, E, th, cs, sn, hr, hi, pr, pi);
    float til = 0.0f;
#pragma unroll
    for (int k = 0; k < NK; ++k) {
      float fr = Fr[k] * hr[k] - Fi[k] * hi[k] + ft * pr[k];
      float fi = Fr[k] * hi[k] + Fi[k] * hr[k] + ft * pi[k];
      Fr[k] = fr; Fi[k] = fi;
      til += er[k] * fr - ei[k] * fi;
    }
    if (tau < NT)
      out[((size_t)(t0 + i) * DCH + (size_t)d) * NT + (size_t)tau] = til * inv_tau;
  }
}

// Fallback when no workspace: direct full-T scan (1 block per d, lane = tau).
__global__ __launch_bounds__(64) void cme_direct(const float* __restrict__ f,
                                                 const float* __restrict__ alpha,
                                                 const float2* __restrict__ F0,
                                                 const float2* __restrict__ s,
                                                 const float2* __restrict__ eta,
                                                 const float* __restrict__ tau_stars,
                                                 float* __restrict__ out) {
  __shared__ float sf[CH_MAX];
  __shared__ float sal[CH_MAX];
  int d = (int)blockIdx.x;
  int tau = (int)threadIdx.x;
  int tt = tau < NT ? tau : NT - 1;
  float sr0 = s[tt * NK + 0].x;
  float si1 = s[tt * NK + 1].y;
  float inv_tau = 1.0f / tau_stars[tt];
  float er[NK], ei[NK], Fr[NK], Fi[NK];
  size_t chain = (size_t)d * NT + (size_t)tt;
#pragma unroll
  for (int k = 0; k < NK; ++k) {
    float2 e = eta[k];
    er[k] = e.x; ei[k] = e.y;
    float2 v = F0[chain * NK + k];
    Fr[k] = v.x; Fi[k] = v.y;
  }
  for (int t0 = 0; t0 < TSTEPS; t0 += CH_MAX) {
    stage_tile(f, alpha, sf, sal, (size_t)t0 * DCH + (size_t)d, CH_MAX);
    for (int i = 0; i < CH_MAX; ++i) {
      float ft = sf[i], at = sal[i];
      float x = at * sr0;
      float th = at * si1;
      float E = __expf(-x);
      float sn, cs;
      __sincosf(th, &sn, &cs);
      float hr[NK], hi[NK], pr[NK], pi[NK];
      step_coeffs(x, E, th, cs, sn, hr, hi, pr, pi);
      float til = 0.0f;
#pragma unroll
      for (int k = 0; k < NK; ++k) {
        float fr = Fr[k] * hr[k] - Fi[k] * hi[k] + ft * pr[k];
        float fi = Fr[k] * hi[k] + Fi[k] * hr[k] + ft * pi[k];
        Fr[k] = fr; Fi[k] = fi;
        til += er[k] * fr - ei[k] * fi;
      }
      if (tau < NT)
        out[((size_t)(t0 + i) * DCH + (size_t)d) * NT + (size_t)tau] = til * inv_tau;
    }
    __syncthreads(); // LDS reuse across outer chunks
  }
}

extern "C" void kernel_launch(void* const* d_in, const int* in_sizes, int n_in,
                              void* d_out, int out_size, void* d_ws, size_t ws_size,
                              hipStream_t stream) {
  (void)in_sizes; (void)n_in; (void)out_size;
  const float* f = (const float*)d_in[0];
  const float* alpha = (const float*)d_in[1];
  const float2* F0 = (const float2*)d_in[2];
  const float2* s = (const float2*)d_in[3];
  const float2* eta = (const float2*)d_in[4];
  const float* tau_stars = (const float*)d_in[5];
  float* out = (float*)d_out;

  // Workspace per chunk-count C: ws1 = 3200*C*40 floats (A,B), ws2 = 3200*C*20 floats (F).
  int C = 32;
  while (C >= 2 && (size_t)NCHAIN * (size_t)C * 60u * sizeof(float) > ws_size) C >>= 1;

  if (C >= 2 && d_ws != nullptr) {
    float* ws1 = (float*)d_ws;
    float* ws2 = ws1 + (size_t)NCHAIN * (size_t)C * 40u;
    int chunkLen = TSTEPS / C;
    dim3 grid((unsigned)DCH, (unsigned)C), block(64);
    hipLaunchKernelGGL(cme_pass1, grid, block, 0, stream, f, alpha, s, ws1, chunkLen);
    hipLaunchKernelGGL(cme_pass2, dim3(NCHAIN / 64), block, 0, stream, F0, ws1, ws2, C);
    hipLaunchKernelGGL(cme_pass3, grid, block, 0, stream, f, alpha, s, eta, tau_stars, ws2, out,
                       chunkLen);
  } else {
    hipLaunchKernelGGL(cme_direct, dim3(DCH), dim3(64), 0, stream, f, alpha, F0, s, eta,
                       tau_stars, out);
  }
}